// PointGroup_84310208020729
// MI455X (gfx1250) — compile-verified
//
#include <hip/hip_runtime.h>

#define B_ 16
#define N_ 8192
#define C_ 6
#define G_ 512
#define K_ 32
#define CHUNK 512
#define NW 16

typedef __attribute__((ext_vector_type(2))) float v2f;
typedef __attribute__((ext_vector_type(8))) float v8f;

// ---------------------------------------------------------------------------
// Kernel 1: Farthest point sampling. One block per batch; xyz resident in LDS
// (96KB of the 320KB WGP LDS); min_d2 in registers (8 pts/thread @ 1024 thr).
// Argmax tie-break = lowest index (matches jnp.argmax).
// ---------------------------------------------------------------------------
__global__ __launch_bounds__(1024) void fps_kernel(const float* __restrict__ in,
                                                   float* __restrict__ centers)
{
    extern __shared__ float smem[];
    float* sx  = smem;               // N_
    float* sy  = smem + N_;          // N_
    float* sz  = smem + 2 * N_;      // N_
    float* rv  = smem + 3 * N_;      // 32 wave-best values
    int*   ri  = (int*)(smem + 3 * N_ + 32); // 32 wave-best indices
    float* cur = smem + 3 * N_ + 64; // 3 floats: current center

    const int b   = blockIdx.x;
    const int tid = threadIdx.x;
    const float* src = in + (size_t)b * N_ * C_;

    for (int p = tid; p < N_; p += 1024) {
        sx[p] = src[p * C_ + 0];
        sy[p] = src[p * C_ + 1];
        sz[p] = src[p * C_ + 2];
    }
    __syncthreads();

    float md[8];
#pragma unroll
    for (int j = 0; j < 8; ++j) md[j] = 3.402823466e38f;

    float cx = sx[0], cy = sy[0], cz = sz[0];
    if (tid == 0) {
        float* c0 = centers + (size_t)b * G_ * 3;
        c0[0] = cx; c0[1] = cy; c0[2] = cz;
    }

    const int lane = tid & 31;
    const int wid  = tid >> 5;

    for (int it = 1; it < G_; ++it) {
        float bv = -1.0f; int bi = N_;
#pragma unroll
        for (int j = 0; j < 8; ++j) {
            int p = tid + j * 1024;
            float dx = sx[p] - cx, dy = sy[p] - cy, dz = sz[p] - cz;
            float d2 = dx * dx + dy * dy + dz * dz;
            float m = fminf(md[j], d2);
            md[j] = m;
            if (m > bv || (m == bv && p < bi)) { bv = m; bi = p; }
        }
        // wave32 argmax reduction, first-index tie-break
#pragma unroll
        for (int off = 16; off > 0; off >>= 1) {
            float ov = __shfl_xor(bv, off);
            int   oi = __shfl_xor(bi, off);
            if (ov > bv || (ov == bv && oi < bi)) { bv = ov; bi = oi; }
        }
        if (lane == 0) { rv[wid] = bv; ri[wid] = bi; }
        __syncthreads();
        if (tid == 0) {
            float best = rv[0]; int besti = ri[0];
            for (int w = 1; w < 32; ++w) {
                float v = rv[w]; int i2 = ri[w];
                if (v > best || (v == best && i2 < besti)) { best = v; besti = i2; }
            }
            float nx = sx[besti], ny = sy[besti], nz = sz[besti];
            cur[0] = nx; cur[1] = ny; cur[2] = nz;
            float* co = centers + ((size_t)b * G_ + it) * 3;
            co[0] = nx; co[1] = ny; co[2] = nz;
        }
        __syncthreads();
        cx = cur[0]; cy = cur[1]; cz = cur[2];
    }
}

// ---------------------------------------------------------------------------
// Kernel 2: kNN(32) of 16 centers vs 8192 points via V_WMMA_F32_16X16X4_F32,
// fused with top-32 selection, 32-lane bitonic sort, and neighborhood gather.
// A-row = (qx,qy,qz,1); B-col = (-2x,-2y,-2z,|x|^2) -> D = |x|^2 - 2 q.x,
// which ranks identically to true d2 per row.
// Block = 512 threads (16 waves); wave w owns center row w.
// Operand construction is branchless (v_cndmask) so the point fetch is a
// single unconditional global_load_b96 and EXEC stays all-ones around WMMA.
// ---------------------------------------------------------------------------
__global__ __launch_bounds__(512) void knn_group_kernel(const float* __restrict__ in,
                                                        const float* __restrict__ centers,
                                                        float* __restrict__ nbh)
{
    __shared__ float d2s[16 * CHUNK];   // 32 KB distance tile strip
    __shared__ float qs[48];            // 16 centers x 3

    const int b    = blockIdx.x >> 5;   // 32 center-tiles per batch
    const int tile = blockIdx.x & 31;
    const int tid  = threadIdx.x;
    const int lane = tid & 31;
    const int wid  = tid >> 5;

    const float* src = in + (size_t)b * N_ * C_;
    const float* ctr = centers + ((size_t)b * G_ + tile * 16) * 3;

    if (tid < 48) qs[tid] = ctr[tid];
    __syncthreads();

    // A operand (16x4 fp32): lanes 0-15 hold K=0,1; lanes 16-31 hold K=2,3
    const int  half = lane >> 4;
    const int  mrow = lane & 15;
    const bool hi   = (half != 0);
    v2f A;
    A.x = hi ? qs[mrow * 3 + 2] : qs[mrow * 3 + 0];
    A.y = hi ? 1.0f             : qs[mrow * 3 + 1];

    // top-32 set: one (value,index) per lane; thr = current worst of set
    float topv = 3.402823466e38f;
    int   topi = 0;
    float thr  = 3.402823466e38f;

    for (int chunk = 0; chunk < N_; chunk += CHUNK) {
        __syncthreads(); // d2s reuse guard
        // --- compute phase: 32 tiles of 16 points; 2 WMMAs per wave ---
#pragma unroll
        for (int tt = 0; tt < 2; ++tt) {
            const int t = wid + tt * NW;
            const int n = chunk + t * 16 + mrow;
            const float* p = src + (size_t)n * C_;
            float x = p[0], y = p[1], z = p[2];
            float n2 = x * x + y * y + z * z;
            v2f Bm;
            Bm.x = hi ? (-2.0f * z) : (-2.0f * x);
            Bm.y = hi ? n2          : (-2.0f * y);
            v8f Cm = {0.f, 0.f, 0.f, 0.f, 0.f, 0.f, 0.f, 0.f};
            v8f D = __builtin_amdgcn_wmma_f32_16x16x4_f32(
                false, A, false, Bm, (short)0, Cm, false, false);
            // D layout: VGPR r, lane l -> row = r + 8*(l>=16), col = l%16
#pragma unroll
            for (int r = 0; r < 8; ++r)
                d2s[(r + 8 * half) * CHUNK + t * 16 + mrow] = D[r];
        }
        __syncthreads();
        // --- selection phase: wave w scans row w ---
        const float* row = d2s + wid * CHUNK;
        for (int c0 = 0; c0 < CHUNK; c0 += 32) {
            float val = row[c0 + lane];
            int   gi  = chunk + c0 + lane;
            unsigned long long m = __ballot(val < thr);
            while (m) {
                int s = __ffsll(m) - 1;
                m &= m - 1;
                float v  = __shfl(val, s);
                int   vi = __shfl(gi, s);
                if (v < thr) {
                    // replace current worst element of the set
                    unsigned long long bm = __ballot(topv == thr);
                    int ml = __ffsll(bm) - 1;
                    if (lane == ml) { topv = v; topi = vi; }
                    float t2 = topv;
#pragma unroll
                    for (int off = 16; off > 0; off >>= 1)
                        t2 = fmaxf(t2, __shfl_xor(t2, off));
                    thr = t2;
                }
            }
        }
    }

    // --- 32-lane bitonic sort, ascending (d2, idx) to match top_k order ---
#pragma unroll
    for (int k = 2; k <= 32; k <<= 1) {
#pragma unroll
        for (int j = k >> 1; j > 0; j >>= 1) {
            float ov = __shfl_xor(topv, j);
            int   oi = __shfl_xor(topi, j);
            bool up      = ((lane & k) == 0);
            bool lower   = ((lane & j) == 0);
            bool takeMin = (lower == up);
            bool otherSmaller = (ov < topv) || (ov == topv && oi < topi);
            if (takeMin == otherSmaller) { topv = ov; topi = oi; }
        }
    }

    // --- gather epilogue: lane j writes the j-th nearest neighbor ---
    const int g = tile * 16 + wid;
    float ccx = qs[wid * 3 + 0], ccy = qs[wid * 3 + 1], ccz = qs[wid * 3 + 2];
    const float* p = src + (size_t)topi * C_;
    float* o = nbh + (((size_t)b * G_ + g) * K_ + lane) * C_;
    o[0] = p[0] - ccx;
    o[1] = p[1] - ccy;
    o[2] = p[2] - ccz;
    o[3] = p[3];
    o[4] = p[4];
    o[5] = p[5];
}

extern "C" void kernel_launch(void* const* d_in, const int* in_sizes, int n_in,
                              void* d_out, int out_size, void* d_ws, size_t ws_size,
                              hipStream_t stream)
{
    (void)in_sizes; (void)n_in; (void)out_size; (void)d_ws; (void)ws_size;
    const float* in = (const float*)d_in[0];
    float* out = (float*)d_out;
    float* nbh     = out;                               // (B,G,K,C)
    float* centers = out + (size_t)B_ * G_ * K_ * C_;   // (B,G,3)

    size_t smem1 = (size_t)(3 * N_ + 80) * sizeof(float);
    fps_kernel<<<B_, 1024, smem1, stream>>>(in, centers);
    knn_group_kernel<<<B_ * 32, 512, 0, stream>>>(in, centers, nbh);
}